// EditableModel_14791867368156
// MI455X (gfx1250) — compile-verified
//
#include <hip/hip_runtime.h>

#define IGNORE_IDX (-100)

constexpr int B = 8, T = 128, V = 50257;
constexpr int NROW = T - 1;                 // 127 predicted positions
constexpr int NV4  = V / 4;                 // 12564 float4s (covers 50256 elems)
constexpr int NST  = (NV4 + 255) / 256;     // 50 stages of 256 float4s per block
constexpr int DEPTH = 4;                    // async stages in flight
constexpr int SLOTS = 8;                    // LDS ring slots (>= 2*DEPTH)

__device__ __forceinline__ void wait_async_le(int n) {
  // n is always a compile-time constant at each call site after unrolling,
  // so this folds to a single s_wait_asynccnt.
  switch (n) {
    case 0: asm volatile("s_wait_asynccnt 0" ::: "memory"); break;
    case 1: asm volatile("s_wait_asynccnt 1" ::: "memory"); break;
    case 2: asm volatile("s_wait_asynccnt 2" ::: "memory"); break;
    case 3: asm volatile("s_wait_asynccnt 3" ::: "memory"); break;
    default: asm volatile("s_wait_asynccnt 4" ::: "memory"); break;
  }
}

// One block per (position i, batch b, tensor which). Streams one 50257-float
// row through a 4-deep async-to-LDS pipeline and produces tok_lp / mask.
__global__ __launch_bounds__(256) void row_lse_kernel(
    const float* __restrict__ pred_anti,
    const float* __restrict__ pred_stereo,
    const int*   __restrict__ targ_anti,
    const int*   __restrict__ targ_stereo,
    float* __restrict__ tok_out,
    float* __restrict__ msk_out)
{
  const int i     = blockIdx.x;   // 0..126
  const int b     = blockIdx.y;   // 0..7
  const int which = blockIdx.z;   // 0=anti, 1=stereo

  const float* pred = which ? pred_stereo : pred_anti;
  const int*   targ = which ? targ_stereo : targ_anti;
  const float* row  = pred + ((size_t)b * T + i) * V;
  const float4* row4 = (const float4*)row;

  __shared__ float4 tiles[SLOTS][256];   // 32 KB ring buffer
  __shared__ float wm[8], wsum[8];

  const int wl = threadIdx.x;            // 0..255 slot id

  // Issue one async non-temporal b128 load per lane for stage S. Index is
  // clamped so every wave issues exactly one instruction per stage (uniform
  // ASYNCcnt pacing, required because completion is in-order and we pace with
  // fixed s_wait_asynccnt counts). Data is single-use (~408 MB >> 192 MB L2),
  // so stream with TH=NT to avoid L2 thrash.
#define ISSUE_STAGE(S) do {                                                   \
    int idx_ = (S) * 256 + wl;                                                \
    idx_ = idx_ < (NV4 - 1) ? idx_ : (NV4 - 1);                               \
    unsigned lds_ = (unsigned)(size_t)&tiles[(S) & (SLOTS - 1)][wl];          \
    const float4* gp_ = row4 + idx_;                                          \
    asm volatile("global_load_async_to_lds_b128 %0, %1, off th:TH_LOAD_NT"    \
                 :: "v"(lds_), "v"(gp_) : "memory");                          \
  } while (0)

  for (int s = 0; s < DEPTH; ++s) ISSUE_STAGE(s);

  // Online logsumexp accumulators (per lane).
  float m = -3.402823466e38f;
  float acc = 0.0f;

  // Steady state: stages [0, NST-DEPTH). All lanes fully in range here
  // (the only partial stage is NST-1), so no validity check in the hot loop.
  for (int s = 0; s < NST - DEPTH; ++s) {
    ISSUE_STAGE(s + DEPTH);
    wait_async_le(DEPTH);                    // 5 outstanding -> oldest done
    float4 v = tiles[s & (SLOTS - 1)][wl];   // ds_load_b128
    float m4 = fmaxf(fmaxf(v.x, v.y), fmaxf(v.z, v.w));
    if (m4 > m) { acc *= __expf(m - m4); m = m4; }   // rare rescale
    acc += __expf(v.x - m) + __expf(v.y - m) +
           __expf(v.z - m) + __expf(v.w - m);
  }

  // Epilogue: last DEPTH stages, constant waits 3,2,1,0; mask partial stage.
#pragma unroll
  for (int k = 0; k < DEPTH; ++k) {
    const int s = NST - DEPTH + k;
    wait_async_le(DEPTH - 1 - k);
    float4 v = tiles[s & (SLOTS - 1)][wl];
    int idx = s * 256 + wl;
    if (idx < NV4) {
      float m4 = fmaxf(fmaxf(v.x, v.y), fmaxf(v.z, v.w));
      if (m4 > m) { acc *= __expf(m - m4); m = m4; }
      acc += __expf(v.x - m) + __expf(v.y - m) +
             __expf(v.z - m) + __expf(v.w - m);
    }
  }
#undef ISSUE_STAGE

  // V % 4 == 1: fold the final scalar element into lane 0's accumulator.
  if (wl == 0) {
    float x = row[V - 1];
    if (x > m) { acc *= __expf(m - x); m = x; }
    acc += __expf(x - m);
  }

  // wave32 cross-lane merge of (m, acc)
  for (int off = 16; off > 0; off >>= 1) {
    float mo = __shfl_xor(m,   off, 32);
    float so = __shfl_xor(acc, off, 32);
    float M  = fmaxf(m, mo);
    acc = acc * __expf(m - M) + so * __expf(mo - M);
    m = M;
  }
  if ((wl & 31) == 0) { wm[wl >> 5] = m; wsum[wl >> 5] = acc; }
  __syncthreads();

  if (wl == 0) {
    float M = wm[0], S = wsum[0];
    for (int w = 1; w < 8; ++w) {
      float mo = wm[w], so = wsum[w];
      float Mn = fmaxf(M, mo);
      S = S * __expf(M - Mn) + so * __expf(mo - Mn);
      M = Mn;
    }
    float lse = M + __logf(S);
    int t = targ[b * T + (i + 1)];
    int r = (which * B + b) * NROW + i;
    if (t != IGNORE_IDX) {
      tok_out[r] = row[t] - lse;   // log_softmax gather
      msk_out[r] = 1.0f;
    } else {
      tok_out[r] = 0.0f;
      msk_out[r] = 0.0f;
    }
  }
}

// Single-wave epilogue: 16 scores, symmetric KL loss, means, ss_score.
__global__ __launch_bounds__(32) void finalize_kernel(
    const float* __restrict__ tok,
    const float* __restrict__ msk,
    float* __restrict__ out)
{
  __shared__ float sc[16];
  const int tid = threadIdx.x;
  if (tid < 16) {                       // tid = which*8 + b
    float s = 0.f, c = 0.f;
    const float* tp = tok + tid * NROW;
    const float* mp = msk + tid * NROW;
    for (int i = 0; i < NROW; ++i) { s += tp[i]; c += mp[i]; }
    sc[tid] = s / c;
  }
  __syncthreads();
  if (tid == 0) {
    float kl1 = 0.f, kl2 = 0.f, am = 0.f, sm = 0.f, ss = 0.f;
    for (int j = 0; j < B; ++j) {
      float a  = sc[j];
      float st = sc[8 + j];
      kl1 += expf(st) * (st - a);
      kl2 += expf(a)  * (a - st);
      am += a; sm += st;
      ss += (st > a) ? 1.0f : 0.0f;
    }
    out[0] = kl1 / (float)B + kl2 / (float)B;   // loss
    for (int j = 0; j < B; ++j) { out[1 + j] = sc[j]; out[9 + j] = sc[8 + j]; }
    out[17] = am / (float)B;                    // anti_score.mean()
    out[18] = sm / (float)B;                    // stereo_score.mean()
    out[19] = ss / (float)B;                    // ss_score
  }
}

extern "C" void kernel_launch(void* const* d_in, const int* in_sizes, int n_in,
                              void* d_out, int out_size, void* d_ws, size_t ws_size,
                              hipStream_t stream) {
  (void)in_sizes; (void)n_in; (void)out_size; (void)ws_size;
  const float* pred_anti   = (const float*)d_in[0];
  const int*   targ_anti   = (const int*)  d_in[1];
  const float* pred_stereo = (const float*)d_in[2];
  const int*   targ_stereo = (const int*)  d_in[3];
  float* out = (float*)d_out;

  float* tok = (float*)d_ws;                 // 2*8*127 floats
  float* msk = tok + 2 * B * NROW;           // 2*8*127 floats

  dim3 grid(NROW, B, 2);
  row_lse_kernel<<<grid, 256, 0, stream>>>(pred_anti, pred_stereo,
                                           targ_anti, targ_stereo, tok, msk);
  finalize_kernel<<<1, 32, 0, stream>>>(tok, msk, out);
}